// SDGCN_24283745091829
// MI455X (gfx1250) — compile-verified
//
#include <hip/hip_runtime.h>
#include <hip/hip_bf16.h>
#include <math.h>

// ---------------------------------------------------------------------------
// SDGCN pipeline for MI455X (gfx1250, wave32).
// GEMMs use V_WMMA_F32_16X16X4_F32 (4 N-tiles per A fragment, k-pair
// interleaved LDS so each B operand is one ds_load_b64);
// sparse passes use wave-per-edge float atomics.
// ---------------------------------------------------------------------------

typedef __attribute__((ext_vector_type(2))) float v2f;
typedef __attribute__((ext_vector_type(8))) float v8f;

#define KDIM 512     // every GEMM in this model has K = 512
#define HIDC 256
#define KCHUNK 128   // K rows of W staged in LDS per stage (128x64 f32 = 32 KB)

// ---------------------------------------------------------------------------
// WMMA f32 GEMM:  C(M x NCOLS) = A(M x 512) * W(512 x NCOLS), optional ReLU.
// Block = 256 threads = 8 waves. Block tile = 128 rows x 64 cols:
// each wave owns 16 rows and computes FOUR 16x16 C tiles from a single A
// fragment per k-step (4x A-operand reuse). W staged in LDS in K-chunks,
// stored k-pair interleaved: wt2[p*64+c] = {W[2p][c], W[2p+1][c]} so a B
// fragment is a single aligned 8B LDS load into an even VGPR pair.
//
// A-operand VGPR layout per ISA (16x4 f32): lanes 0-15 -> M rows, {K,K+1};
// lanes 16-31 -> same M rows, {K+2,K+3}.  B mirrors with N across lanes.
// EXEC must be all ones for WMMA: out-of-range rows are clamped on load and
// masked on store only.
// ---------------------------------------------------------------------------
template <int NCOLS, bool RELU>
__global__ __launch_bounds__(256) void gemm_wmma_k512(
    const float* __restrict__ A, const float* __restrict__ W,
    float* __restrict__ C, int M)
{
    __shared__ float2 wt2[(KCHUNK / 2) * 64];

    const int tid     = threadIdx.x;
    const int colBase = blockIdx.y * 64;

    const int lane  = tid & 31;
    const int wave  = tid >> 5;
    const int row0  = blockIdx.x * 128 + wave * 16;
    const int mrow  = row0 + (lane & 15);
    const int rcl   = (mrow < M) ? mrow : (M - 1);       // clamp, keep EXEC full
    const float* __restrict__ Arow = A + (size_t)rcl * KDIM;
    const int khalf = (lane >> 4) << 1;                  // 0 for lanes<16, 2 else
    const int col   = lane & 15;

    v8f acc0 = {}, acc1 = {}, acc2 = {}, acc3 = {};

    for (int kc = 0; kc < KDIM; kc += KCHUNK) {
        __syncthreads();   // WAR guard before overwriting the staged chunk
        // Cooperative stage: k-pair interleaved (KCHUNK/2) x 64 float2 tile.
        for (int j = tid; j < (KCHUNK / 2) * 64; j += 256) {
            const int p = j >> 6;
            const int c = j & 63;
            const float* __restrict__ wp = &W[(kc + 2 * p) * NCOLS + colBase + c];
            wt2[j] = make_float2(wp[0], wp[NCOLS]);
        }
        __syncthreads();

#pragma unroll 8
        for (int k = 0; k < KCHUNK; k += 4) {
            const v2f a = *(const v2f*)(Arow + kc + k + khalf);          // 8B aligned
            const v2f* __restrict__ bp =
                (const v2f*)&wt2[((k + khalf) >> 1) * 64 + col];         // even pair idx
            acc0 = __builtin_amdgcn_wmma_f32_16x16x4_f32(false, a, false, bp[0],
                                                         (short)0, acc0, false, false);
            acc1 = __builtin_amdgcn_wmma_f32_16x16x4_f32(false, a, false, bp[16],
                                                         (short)0, acc1, false, false);
            acc2 = __builtin_amdgcn_wmma_f32_16x16x4_f32(false, a, false, bp[32],
                                                         (short)0, acc2, false, false);
            acc3 = __builtin_amdgcn_wmma_f32_16x16x4_f32(false, a, false, bp[48],
                                                         (short)0, acc3, false, false);
        }
    }

    const int half = lane >> 4;   // C/D layout: lanes16-31 hold M = i+8
    v8f* accs[4] = {&acc0, &acc1, &acc2, &acc3};
#pragma unroll
    for (int t = 0; t < 4; ++t) {
        const int colg = colBase + t * 16 + col;
#pragma unroll
        for (int i = 0; i < 8; ++i) {
            const int r = row0 + i + half * 8;
            if (r < M) {
                float v = (*accs[t])[i];
                if (RELU) v = v > 0.f ? v : 0.f;
                C[(size_t)r * NCOLS + colg] = v;
            }
        }
    }
}

// ---------------------------------------------------------------------------
// Small elementwise / graph kernels.
// ---------------------------------------------------------------------------
__global__ void zero_kernel(float* __restrict__ p, size_t n) {
    size_t i = (size_t)blockIdx.x * blockDim.x + threadIdx.x;
    if (i < n) p[i] = 0.f;
}

// deg[i] = in-degree(dst); cnt[i] = out-degree(src)
__global__ void deg_cnt_kernel(const int* __restrict__ src, const int* __restrict__ dst,
                               float* __restrict__ deg, float* __restrict__ cnt, int E) {
    int e = blockIdx.x * blockDim.x + threadIdx.x;
    if (e < E) {
        atomicAdd(&deg[dst[e]], 1.0f);
        atomicAdd(&cnt[src[e]], 1.0f);
    }
}

// dinv = rsqrt(deg + 1 /*self loop*/);  cnt = max(cnt, 1)
__global__ void deg_final_kernel(float* __restrict__ dinv, float* __restrict__ cnt, int n) {
    int i = blockIdx.x * blockDim.x + threadIdx.x;
    if (i < n) {
        dinv[i] = rsqrtf(dinv[i] + 1.0f);
        cnt[i]  = fmaxf(cnt[i], 1.0f);
    }
}

// out[i,c] = h[i,c] * dinv[i]^2   (self-loop contribution of GCN aggregation)
__global__ void agg_init_kernel(const float* __restrict__ h, const float* __restrict__ dinv,
                                float* __restrict__ out, int n, int C) {
    size_t i = (size_t)blockIdx.x * blockDim.x + threadIdx.x;
    if (i < (size_t)n * C) {
        const int row = (int)(i / C);
        const float d = dinv[row];
        out[i] = h[i] * d * d;
    }
}

// wave-per-edge: out[dst] += h[src] * dinv[src]*dinv[dst]
// channel map c = lane + 32*j -> every gather/atomic instruction covers a
// dense 128B line across the wave.
__global__ void agg_edges_kernel(const float* __restrict__ h, const int* __restrict__ src,
                                 const int* __restrict__ dst, const float* __restrict__ dinv,
                                 float* __restrict__ out, int E, int C) {
    const int lane  = threadIdx.x & 31;
    const int gwave = (blockIdx.x * blockDim.x + threadIdx.x) >> 5;
    if (gwave >= E) return;
    const int s = src[gwave];
    const int d = dst[gwave];
    const float coef = dinv[s] * dinv[d];
    const float* __restrict__ hs = h + (size_t)s * C;
    float* __restrict__ od = out + (size_t)d * C;
    for (int c = lane; c < C; c += 32)
        atomicAdd(&od[c], hs[c] * coef);
}

template <bool RELU>
__global__ void agg_final_kernel(float* __restrict__ out, const float* __restrict__ bias,
                                 int n, int C) {
    size_t i = (size_t)blockIdx.x * blockDim.x + threadIdx.x;
    if (i < (size_t)n * C) {
        float v = out[i] + bias[(int)(i % C)];
        if (RELU) v = v > 0.f ? v : 0.f;
        out[i] = v;
    }
}

// wave-per-edge spmm-sum: out[src] += v[dst]  (ABSDIFF: |v[dst]-vm[dst]| fused)
template <bool ABSDIFF>
__global__ void spmm_edges_kernel(const float* __restrict__ v, const float* __restrict__ vm,
                                  const int* __restrict__ src, const int* __restrict__ dst,
                                  float* __restrict__ out, int E, int C) {
    const int lane  = threadIdx.x & 31;
    const int gwave = (blockIdx.x * blockDim.x + threadIdx.x) >> 5;
    if (gwave >= E) return;
    const int s = src[gwave];
    const int d = dst[gwave];
    const float* __restrict__ vd = v + (size_t)d * C;
    const float* __restrict__ md = ABSDIFF ? (vm + (size_t)d * C) : nullptr;
    float* __restrict__ os = out + (size_t)s * C;
    for (int c = lane; c < C; c += 32) {
        float val = vd[c];
        if (ABSDIFF) val = fabsf(val - md[c]);
        atomicAdd(&os[c], val);
    }
}

__global__ void spmm_final_kernel(float* __restrict__ out, const float* __restrict__ cnt,
                                  int n, int C) {
    size_t i = (size_t)blockIdx.x * blockDim.x + threadIdx.x;
    if (i < (size_t)n * C) out[i] /= cnt[(int)(i / C)];
}

// ---------------------------------------------------------------------------
// WeightScoreLayer + de-smoothing fusion + l2norm. One wave32 per node.
// score = sigmoid( sum_c hm*h*Wsc[c] + hstd*Wsc[256+c] + h*Wsc[512+c] )
// hc[i] = l2norm( concat[(1-score)*h, score*x0] )   (512-dim norm)
// ---------------------------------------------------------------------------
__global__ void score_fuse_kernel(const float* __restrict__ h, const float* __restrict__ hm,
                                  const float* __restrict__ hs, const float* __restrict__ x0,
                                  const float* __restrict__ wsc, float* __restrict__ hc, int n) {
    const int lane = threadIdx.x & 31;
    const int node = (blockIdx.x * blockDim.x + threadIdx.x) >> 5;
    if (node >= n) return;
    const float* __restrict__ H  = h  + (size_t)node * HIDC;
    const float* __restrict__ HM = hm + (size_t)node * HIDC;
    const float* __restrict__ HS = hs + (size_t)node * HIDC;
    const float* __restrict__ X  = x0 + (size_t)node * HIDC;

    float hh[8], xx[8];
    float logit = 0.f;
#pragma unroll
    for (int j = 0; j < 8; ++j) {
        const int c  = lane + 32 * j;
        const float hv = H[c];
        hh[j] = hv;
        xx[j] = X[c];
        logit += HM[c] * hv * wsc[c] + HS[c] * wsc[HIDC + c] + hv * wsc[2 * HIDC + c];
    }
#pragma unroll
    for (int off = 16; off > 0; off >>= 1) logit += __shfl_xor(logit, off, 32);

    const float score = 1.0f / (1.0f + expf(-logit));
    const float wa = 1.0f - score, wb = score;

    float ss = 0.f;
#pragma unroll
    for (int j = 0; j < 8; ++j) {
        const float l = wa * hh[j], r = wb * xx[j];
        ss += l * l + r * r;
    }
#pragma unroll
    for (int off = 16; off > 0; off >>= 1) ss += __shfl_xor(ss, off, 32);

    const float inv = 1.0f / fmaxf(sqrtf(ss), 1e-12f);
    float* __restrict__ O = hc + (size_t)node * (2 * HIDC);
#pragma unroll
    for (int j = 0; j < 8; ++j) {
        const int c = lane + 32 * j;
        O[c]        = wa * hh[j] * inv;
        O[HIDC + c] = wb * xx[j] * inv;
    }
}

// ---------------------------------------------------------------------------
// Host-side orchestration.
// ---------------------------------------------------------------------------
static inline int cdiv(long long a, long long b) { return (int)((a + b - 1) / b); }

extern "C" void kernel_launch(void* const* d_in, const int* in_sizes, int n_in,
                              void* d_out, int out_size, void* d_ws, size_t ws_size,
                              hipStream_t stream) {
    const float* x    = (const float*)d_in[0];
    const int*   ei   = (const int*)d_in[1];
    const float* W0   = (const float*)d_in[2];
    const float* b0   = (const float*)d_in[3];
    const float* W1   = (const float*)d_in[4];
    const float* b1   = (const float*)d_in[5];
    const float* W2   = (const float*)d_in[6];
    const float* b2   = (const float*)d_in[7];
    const float* Wres = (const float*)d_in[8];
    const float* Wsc  = (const float*)d_in[9];

    const int N = in_sizes[0] / KDIM;      // 50000
    const int E = in_sizes[1] / 2;         // 800000
    const int OUTC = in_sizes[7];          // 64
    const int* src = ei;
    const int* dst = ei + E;

    float* ws = (float*)d_ws;
    const size_t NH = (size_t)N * HIDC;
    float* X0   = ws;                       // N x 256  relu(x@Wres)
    float* HL   = X0 + NH;                  // N x 256  linear out / hstd (reused)
    float* HA   = HL + NH;                  // N x 256  conv output (post relu)
    float* HM   = HA + NH;                  // N x 256  h_mean
    float* HC   = HM + NH;                  // N x 512  fused concat
    float* DINV = HC + (size_t)N * 2 * HIDC;
    float* CNT  = DINV + N;

    const dim3 blk(256);
    const int eBlocks  = cdiv((long long)E * 32, 256);       // wave-per-edge
    const int nWaveBlk = cdiv((long long)N * 32, 256);       // wave-per-node
    const int ewBlocks = cdiv(E, 256);
    const int nBlocks  = cdiv(N, 256);
    const int nc256    = cdiv((long long)N * HIDC, 256);
    const int ncOut    = cdiv((long long)N * OUTC, 256);
    const dim3 gemmG256(cdiv(N, 128), HIDC / 64);            // 64 cols per block
    const dim3 gemmGOut(cdiv(N, 128), OUTC / 64);

    // ---- degrees / counts (graph-constant) ----
    zero_kernel<<<nBlocks, blk, 0, stream>>>(DINV, (size_t)N);
    zero_kernel<<<nBlocks, blk, 0, stream>>>(CNT, (size_t)N);
    deg_cnt_kernel<<<ewBlocks, blk, 0, stream>>>(src, dst, DINV, CNT, E);
    deg_final_kernel<<<nBlocks, blk, 0, stream>>>(DINV, CNT, N);

    // ---- x0 = relu(x @ Wres) ----
    gemm_wmma_k512<HIDC, true><<<gemmG256, blk, 0, stream>>>(x, Wres, X0, N);

    const float* hin = x;            // input to current layer's linear
    const float* Wlay[2] = {W0, W1};
    const float* blay[2] = {b0, b1};

    for (int layer = 0; layer < 2; ++layer) {
        // linear: HL = hin @ W
        gemm_wmma_k512<HIDC, false><<<gemmG256, blk, 0, stream>>>(hin, Wlay[layer], HL, N);
        // GCN aggregate -> HA (self loop + edges + bias + relu)
        agg_init_kernel<<<nc256, blk, 0, stream>>>(HL, DINV, HA, N, HIDC);
        agg_edges_kernel<<<eBlocks, blk, 0, stream>>>(HL, src, dst, DINV, HA, E, HIDC);
        agg_final_kernel<true><<<nc256, blk, 0, stream>>>(HA, blay[layer], N, HIDC);
        // h_mean -> HM
        zero_kernel<<<nc256, blk, 0, stream>>>(HM, NH);
        spmm_edges_kernel<false><<<eBlocks, blk, 0, stream>>>(HA, nullptr, src, dst, HM, E, HIDC);
        spmm_final_kernel<<<nc256, blk, 0, stream>>>(HM, CNT, N, HIDC);
        // h_std -> HL (HL is free now; |h - hm| fused into gather)
        zero_kernel<<<nc256, blk, 0, stream>>>(HL, NH);
        spmm_edges_kernel<true><<<eBlocks, blk, 0, stream>>>(HA, HM, src, dst, HL, E, HIDC);
        spmm_final_kernel<<<nc256, blk, 0, stream>>>(HL, CNT, N, HIDC);
        // score + fuse + l2norm -> HC (N x 512)
        score_fuse_kernel<<<nWaveBlk, blk, 0, stream>>>(HA, HM, HL, X0, Wsc, HC, N);
        hin = HC;
    }

    // ---- final layer: out = aggregate(HC @ W2) + b2 (no relu) ----
    float* out = (float*)d_out;
    gemm_wmma_k512<64, false><<<gemmGOut, blk, 0, stream>>>(HC, W2, HL, N);
    agg_init_kernel<<<ncOut, blk, 0, stream>>>(HL, DINV, out, N, OUTC);
    agg_edges_kernel<<<eBlocks, blk, 0, stream>>>(HL, src, dst, DINV, out, E, OUTC);
    agg_final_kernel<false><<<ncOut, blk, 0, stream>>>(out, b2, N, OUTC);
}